// DeepNet_83064667504982
// MI455X (gfx1250) — compile-verified
//
#include <hip/hip_runtime.h>

// ---------------------------------------------------------------------------
// GraphSAGE-style network on gfx1250 (MI455X).
// Memory-bound: ~5GB of edge scatter traffic dominates (~250us @ 23.3 TB/s).
// GEMMs (~23 GFLOP total) run on V_WMMA_F32_16X16X4_F32 (full f32 precision,
// flops are free at this arithmetic intensity). Each layer fuses
// GEMM + bias + graph segment-sum + residual + LayerNorm + ReLU in one kernel.
// ---------------------------------------------------------------------------

typedef float v2f __attribute__((ext_vector_type(2)));
typedef float v8f __attribute__((ext_vector_type(8)));

constexpr int DIM    = 128;   // feature dim
constexpr int NGRAPH = 64;    // graphs
constexpr int NLAYER = 3;

__device__ __forceinline__ v8f wmma_f32(v2f a, v2f b, v8f c) {
  // (neg_a, A, neg_b, B, c_mod, C, reuse_a, reuse_b)
  return __builtin_amdgcn_wmma_f32_16x16x4_f32(false, a, false, b, (short)0, c,
                                               false, false);
}

__device__ __forceinline__ void atomAddF(float* p, float v) {
  unsafeAtomicAdd(p, v);  // hardware global_atomic_add_f32
}

// ---------------------------------------------------------------- degree ----
__global__ void deg_kernel(const long long* __restrict__ dst,
                           float* __restrict__ deg, long long E) {
  long long e = (long long)blockIdx.x * blockDim.x + threadIdx.x;
  if (e < E) atomAddF(&deg[dst[e]], 1.0f);
}

// --------------------------------------------------------- edge scatter ----
// 32 threads (one wave) per edge: coalesced 512B row gather, 4 f32 atomics/lane.
__global__ __launch_bounds__(256) void scatter_kernel(
    const long long* __restrict__ src, const long long* __restrict__ dst,
    const float* __restrict__ node, float* __restrict__ agg, long long E) {
  long long t = (long long)blockIdx.x * 256 + threadIdx.x;
  long long e = t >> 5;
  if (e >= E) return;
  int c = (int)(t & 31) * 4;
  long long s = src[e], d = dst[e];
  const float4 v = *(const float4*)(node + s * DIM + c);
  float* p = agg + d * DIM + c;
  atomAddF(p + 0, v.x);
  atomAddF(p + 1, v.y);
  atomAddF(p + 2, v.z);
  atomAddF(p + 3, v.w);
}

// ----------------------------------------------------- input projection ----
// out[N,128] = x @ W + bias.  One block = 16 rows, 8 waves = 8 column tiles.
__global__ __launch_bounds__(256) void fc_gemm(
    const float* __restrict__ x, const float* __restrict__ W,
    const float* __restrict__ bias, float* __restrict__ out, int N) {
  const int lane = threadIdx.x & 31;
  const int wave = threadIdx.x >> 5;
  const int r0 = blockIdx.x * 16;
  const int c0 = wave * 16;
  const int mrow = lane & 15;
  int row = r0 + mrow;
  if (row >= N) row = N - 1;
  const int kh = (lane < 16) ? 0 : 2;  // A frag: lanes 0-15 K=0,1; 16-31 K=2,3
  const float* xr = x + (long long)row * DIM;
  v8f acc = {};
  for (int kb = 0; kb < DIM; kb += 4) {
    v2f a; a.x = xr[kb + kh];               a.y = xr[kb + kh + 1];
    v2f b; b.x = W[(kb + kh) * DIM + c0 + mrow];
           b.y = W[(kb + kh + 1) * DIM + c0 + mrow];
    acc = wmma_f32(a, b, acc);
  }
  const float bi = bias[c0 + mrow];
#pragma unroll
  for (int r = 0; r < 8; ++r) {
    int m = (lane < 16) ? r : r + 8;  // C/D: VGPR r -> M=r (lo lanes) / r+8
    int orow = r0 + m;
    if (orow < N) out[(long long)orow * DIM + c0 + mrow] = acc[r] + bi;
  }
}

// --------------------------------------------------------- fused layer -----
// h = node@Wself + (agg/deg)@Wneigh + b ; gacc[batch] += h ;
// node_out = relu(LN(h + node)).
__global__ __launch_bounds__(256) void sage_layer(
    const float* __restrict__ node_in, const float* __restrict__ agg,
    const float* __restrict__ deg, const float* __restrict__ Wself,
    const float* __restrict__ Wneigh, const float* __restrict__ cbias,
    const float* __restrict__ lng, const float* __restrict__ lnb,
    const long long* __restrict__ batch, float* __restrict__ gacc,
    float* __restrict__ node_out, int N) {
  __shared__ float hs[16][DIM + 4];
  __shared__ float red_s[16][16];
  __shared__ float red_ss[16][16];
  __shared__ float s_mu[16], s_rs[16];

  const int t = threadIdx.x;
  const int lane = t & 31;
  const int wave = t >> 5;
  const int r0 = blockIdx.x * 16;
  const int c0 = wave * 16;
  const int mrow = lane & 15;
  int row = r0 + mrow;
  if (row >= N) row = N - 1;
  const int kh = (lane < 16) ? 0 : 2;
  const float inv = 1.0f / fmaxf(deg[row], 1.0f);
  const float* nr = node_in + (long long)row * DIM;
  const float* ar = agg + (long long)row * DIM;

  v8f acc = {};
  for (int kb = 0; kb < DIM; kb += 4) {
    v2f a1; a1.x = nr[kb + kh];       a1.y = nr[kb + kh + 1];
    v2f b1; b1.x = Wself[(kb + kh) * DIM + c0 + mrow];
            b1.y = Wself[(kb + kh + 1) * DIM + c0 + mrow];
    acc = wmma_f32(a1, b1, acc);
    v2f a2; a2.x = ar[kb + kh] * inv; a2.y = ar[kb + kh + 1] * inv;
    v2f b2; b2.x = Wneigh[(kb + kh) * DIM + c0 + mrow];
            b2.y = Wneigh[(kb + kh + 1) * DIM + c0 + mrow];
    acc = wmma_f32(a2, b2, acc);
  }
  const float bi = cbias[c0 + mrow];
#pragma unroll
  for (int r = 0; r < 8; ++r) {
    int m = (lane < 16) ? r : r + 8;
    hs[m][c0 + mrow] = acc[r] + bi;  // pre-residual h strip in LDS
  }
  __syncthreads();

  // graph segment-sum (batch is sorted -> run-length compress, ~1 atomic/col)
  if (t < DIM) {
    float a = 0.0f;
    long long cur = batch[(r0 < N) ? r0 : (N - 1)];
    for (int m = 0; m < 16; ++m) {
      int rr = r0 + m;
      if (rr >= N) break;
      long long bm = batch[rr];
      if (bm != cur) {
        atomAddF(&gacc[cur * DIM + t], a);
        a = 0.0f;
        cur = bm;
      }
      a += hs[m][t];
    }
    atomAddF(&gacc[cur * DIM + t], a);
  }

  // residual + LayerNorm + ReLU: thread t -> row t>>4, 8 columns
  const int m = t >> 4;
  const int j = t & 15;
  const int nb = j * 8;
  const int orow = r0 + m;
  const int crow = (orow < N) ? orow : (N - 1);
  float v[8];
  {
    const float* nin = node_in + (long long)crow * DIM + nb;
    float s = 0.0f, ss = 0.0f;
#pragma unroll
    for (int k = 0; k < 8; ++k) {
      float hv = hs[m][nb + k] + nin[k];
      v[k] = hv;
      s += hv;
      ss += hv * hv;
    }
    red_s[m][j] = s;
    red_ss[m][j] = ss;
  }
  __syncthreads();
  if (j == 0) {
    float s = 0.0f, ss = 0.0f;
#pragma unroll
    for (int q = 0; q < 16; ++q) { s += red_s[m][q]; ss += red_ss[m][q]; }
    float mu = s * (1.0f / DIM);
    float var = ss * (1.0f / DIM) - mu * mu;
    s_mu[m] = mu;
    s_rs[m] = rsqrtf(var + 1e-5f);
  }
  __syncthreads();
  if (orow < N) {
    const float mu = s_mu[m], rs = s_rs[m];
    float* op = node_out + (long long)orow * DIM + nb;
#pragma unroll
    for (int k = 0; k < 8; ++k) {
      float y = (v[k] - mu) * rs * lng[nb + k] + lnb[nb + k];
      op[k] = fmaxf(y, 0.0f);
    }
  }
}

// ----------------------------------------------- graph state: g -> LN(LN(g))
__global__ __launch_bounds__(128) void graph_update(
    const float* __restrict__ gacc, const float* __restrict__ gprev,
    const float* __restrict__ lng, const float* __restrict__ lnb,
    float* __restrict__ gout) {
  __shared__ float red[128];
  __shared__ float red2[128];
  const int g = blockIdx.x;
  const int t = threadIdx.x;
  float v = gacc[g * DIM + t] + gprev[g * DIM + t];
  const float gw = lng[t], gb = lnb[t];
#pragma unroll
  for (int pass = 0; pass < 2; ++pass) {
    red[t] = v;
    red2[t] = v * v;
    __syncthreads();
    for (int off = 64; off > 0; off >>= 1) {
      if (t < off) { red[t] += red[t + off]; red2[t] += red2[t + off]; }
      __syncthreads();
    }
    float mu = red[0] * (1.0f / DIM);
    float var = red2[0] * (1.0f / DIM) - mu * mu;
    float rs = rsqrtf(var + 1e-5f);
    __syncthreads();  // before red[] is overwritten next pass
    v = (v - mu) * rs * gw + gb;
  }
  gout[g * DIM + t] = v;
}

// ---------------------------------------------------------------------------
extern "C" void kernel_launch(void* const* d_in, const int* in_sizes, int n_in,
                              void* d_out, int out_size, void* d_ws,
                              size_t ws_size, hipStream_t stream) {
  const float* x        = (const float*)d_in[0];
  const long long* ei   = (const long long*)d_in[1];
  const long long* batch = (const long long*)d_in[2];
  const float* fc_w     = (const float*)d_in[3];
  const float* fc_b     = (const float*)d_in[4];
  const float* w_self   = (const float*)d_in[5];
  const float* w_neigh  = (const float*)d_in[6];
  const float* conv_b   = (const float*)d_in[7];
  const float* ln_g     = (const float*)d_in[8];
  const float* ln_b     = (const float*)d_in[9];

  const int N = in_sizes[2];
  const long long E = (long long)in_sizes[1] / 2;
  const long long ND = (long long)N * DIM;

  float* out_node  = (float*)d_out;
  float* out_graph = out_node + ND;

  // workspace layout
  float* ws     = (float*)d_ws;
  float* node0  = ws;                      // N*D  (ping-pong w/ d_out node)
  float* aggb   = ws + ND;                 // N*D
  float* degb   = ws + 2 * ND;             // N
  float* gaccb  = degb + N;                // G*D
  float* graphb = gaccb + NGRAPH * DIM;    // G*D

  const long long* srcp = ei;
  const long long* dstp = ei + E;

  hipMemsetAsync(degb, 0, (size_t)N * sizeof(float), stream);
  hipMemsetAsync(graphb, 0, (size_t)NGRAPH * DIM * sizeof(float), stream);

  deg_kernel<<<(int)((E + 255) / 256), 256, 0, stream>>>(dstp, degb, E);

  const int row_tiles = (N + 15) / 16;
  fc_gemm<<<row_tiles, 256, 0, stream>>>(x, fc_w, fc_b, node0, N);

  const float* nin = node0;
  float* nout = out_node;
  for (int l = 0; l < NLAYER; ++l) {
    hipMemsetAsync(aggb, 0, (size_t)ND * sizeof(float), stream);
    hipMemsetAsync(gaccb, 0, (size_t)NGRAPH * DIM * sizeof(float), stream);

    long long sthreads = E * 32;
    scatter_kernel<<<(int)((sthreads + 255) / 256), 256, 0, stream>>>(
        srcp, dstp, nin, aggb, E);

    sage_layer<<<row_tiles, 256, 0, stream>>>(
        nin, aggb, degb, w_self + (long long)l * DIM * DIM,
        w_neigh + (long long)l * DIM * DIM, conv_b + l * DIM, ln_g + l * DIM,
        ln_b + l * DIM, batch, gaccb, nout, N);

    float* gout = (l == NLAYER - 1) ? out_graph : graphb;
    graph_update<<<NGRAPH, 128, 0, stream>>>(gaccb, graphb, ln_g + l * DIM,
                                             ln_b + l * DIM, gout);

    const float* tmp = nin;  // ping-pong: layer 2 lands in d_out node region
    nin = nout;
    nout = (float*)tmp;
  }
}